// Model_70523363000700
// MI455X (gfx1250) — compile-verified
//
#include <hip/hip_runtime.h>

// MI455X / gfx1250, wave32. D = A(16x4,f32) * B(4x16,f32) + C(16x16,f32)
typedef float v2f __attribute__((ext_vector_type(2)));
typedef float v4f __attribute__((ext_vector_type(4)));
typedef float v8f __attribute__((ext_vector_type(8)));

#define KNBR 8

// One wave computes one 16-node row tile. 8 waves per 256-thread block.
// W slab for the current neighbor (FIN x FOUT) staged in LDS, K-pair
// interleaved (float2 per (pair,col)) + XOR-16 column swizzle on the
// upper-half pair group so the two half-waves hit disjoint bank groups.
//
// K-permutation (A and B agree): WMMA chunk kc covers global K rows
// {2kc, 2kc+1} + half*(FIN/2). This makes each lane's A data for two
// consecutive kc a contiguous float4.
template <int FIN, int FOUT, bool LEAKY>
__global__ __launch_bounds__(256) void gnn_layer_wmma(
    const float* __restrict__ Hin,   // [N, FIN]
    const float* __restrict__ pos,   // [N, 3]
    const int*   __restrict__ nbr,   // [N, KNBR]
    const float* __restrict__ W,     // [KNBR*FIN, FOUT] row-major
    const float* __restrict__ Bias,  // [FOUT]
    float* __restrict__ Hout,        // [N, FOUT]
    int N)
{
    constexpr int NT = FOUT / 16;    // 16-col output tiles
    constexpr int NP = FIN / 2;      // K-pairs per slab
    constexpr int HP = FIN / 4;      // pairs owned by each half-wave group
    __shared__ v2f Wslab[NP * FOUT]; // <= 64 KB

    const int lane = threadIdx.x & 31;
    const int wave = threadIdx.x >> 5;
    const int tile = blockIdx.x * 8 + wave;  // 16-row tile index
    const int m    = lane & 15;              // row within tile
    const int half = lane >> 4;              // which K half this lane feeds

    int row = tile * 16 + m;
    if (row >= N) row = N - 1;               // clamp; keep EXEC uniform for WMMA

    // Per-row neighbor ids + 1/dist (reference: dist==0 -> 0.5 -> inv = 2.0)
    int   nb[KNBR];
    float invd[KNBR];
    const float px = pos[row * 3 + 0];
    const float py = pos[row * 3 + 1];
    const float pz = pos[row * 3 + 2];
#pragma unroll
    for (int j = 0; j < KNBR; ++j) {
        const int nn = nbr[row * KNBR + j];
        nb[j] = nn;
        const float dx = px - pos[nn * 3 + 0];
        const float dy = py - pos[nn * 3 + 1];
        const float dz = pz - pos[nn * 3 + 2];
        const float d  = sqrtf(dx * dx + dy * dy + dz * dz);
        invd[j] = (d == 0.0f) ? 2.0f : (1.0f / d);
    }

    // Init accumulators with bias (C/D layout: VGPR v -> M = v + 8*half, N = m)
    v8f acc[NT];
#pragma unroll
    for (int t = 0; t < NT; ++t) {
        const float bv = Bias[t * 16 + m];
#pragma unroll
        for (int v = 0; v < 8; ++v) acc[t][v] = bv;
    }

    const int s = half * 16;  // column swizzle for this lane's pair group

    for (int j = 0; j < KNBR; ++j) {
        __syncthreads();
        // Stage slab: pair p holds W rows {2p, 2p+1}; column n of pair p is
        // stored at column (n ^ (p>=HP ? 16 : 0)) & (FOUT-1).
        {
            constexpr int NELT = NP * (FOUT / 2);  // float4 elements
            const float* base = W + (size_t)j * FIN * FOUT;
            for (int e = threadIdx.x; e < NELT; e += 256) {
                const int p = e / (FOUT / 2);
                const int n = (e % (FOUT / 2)) * 2;
                const float* g = base + (size_t)(2 * p) * FOUT + n;
                v2f x = *(const v2f*)g;          // row 2p,   cols n..n+1
                v2f y = *(const v2f*)(g + FOUT); // row 2p+1, cols n..n+1
                const int ss = (p >= HP) ? 16 : 0;
                const int n2 = (n ^ ss) & (FOUT - 1);
                v4f pk = {x.x, y.x, x.y, y.y};
                *(v4f*)&Wslab[p * FOUT + n2] = pk;
            }
        }
        // Prefetch next neighbor's slab into cache while we compute.
        if (j + 1 < KNBR) {
            constexpr int STRIDE = (FIN * FOUT) / 256;  // floats per thread
            __builtin_prefetch(W + (size_t)(j + 1) * FIN * FOUT +
                               (size_t)threadIdx.x * STRIDE, 0, 3);
        }
        __syncthreads();

        const float* arow = Hin + (size_t)nb[j] * FIN;  // gathered neighbor row
        const float  sc   = invd[j];

#pragma unroll
        for (int q = 0; q < FIN / 8; ++q) {
            // One b128 gather feeds 2 kc chunks (= 2*NT WMMAs)
            v4f a4 = *(const v4f*)(arow + 4 * q + half * (FIN / 2));
            v2f a01 = {a4.x * sc, a4.y * sc};
            v2f a23 = {a4.z * sc, a4.w * sc};
#pragma unroll
            for (int kk = 0; kk < 2; ++kk) {
                const v2f a = kk ? a23 : a01;
                const int p = (2 * q + kk) + half * HP;
                const v2f* brow = &Wslab[p * FOUT];
#pragma unroll
                for (int t = 0; t < NT; ++t) {
                    const int n2 = ((t * 16 + m) ^ s) & (FOUT - 1);
                    v2f b = brow[n2];  // ds_load_b64, conflict-free
                    acc[t] = __builtin_amdgcn_wmma_f32_16x16x4_f32(
                        /*neg_a=*/false, a, /*neg_b=*/false, b,
                        /*c_mod=*/(short)0, acc[t],
                        /*reuse_a=*/false, /*reuse_b=*/false);
                }
            }
        }
    }

    // Store D: VGPR v -> global row tile*16 + v + 8*half, col t*16 + m
#pragma unroll
    for (int t = 0; t < NT; ++t) {
#pragma unroll
        for (int v = 0; v < 8; ++v) {
            float o = acc[t][v];
            if (LEAKY) o = (o >= 0.0f) ? o : 0.01f * o;
            const int r = tile * 16 + half * 8 + v;
            if (r < N) Hout[(size_t)r * FOUT + t * 16 + m] = o;
        }
    }
}

extern "C" void kernel_launch(void* const* d_in, const int* in_sizes, int n_in,
                              void* d_out, int out_size, void* d_ws, size_t ws_size,
                              hipStream_t stream) {
    const float* h   = (const float*)d_in[0];  // [N,16]
    const float* pos = (const float*)d_in[1];  // [N,3]
    const int*   nbr = (const int*)d_in[2];    // [N,8]
    const float* W0  = (const float*)d_in[3];  // [128,128]
    const float* b0  = (const float*)d_in[4];  // [128]
    const float* W1  = (const float*)d_in[5];  // [1024,128]
    const float* b1  = (const float*)d_in[6];  // [128]
    const float* W2  = (const float*)d_in[7];  // [1024,16]
    const float* b2  = (const float*)d_in[8];  // [16]
    float*       out = (float*)d_out;          // [N,16]

    const int N = in_sizes[2] / KNBR;          // nbr_idx has N*K elements

    // Intermediates h1,h2: [N,128] f32 each in workspace
    float* h1 = (float*)d_ws;
    float* h2 = h1 + (size_t)N * 128;

    const int tiles  = (N + 15) / 16;
    const int blocks = (tiles + 7) / 8;

    gnn_layer_wmma<16, 128, false><<<blocks, 256, 0, stream>>>(h,  pos, nbr, W0, b0, h1, N);
    gnn_layer_wmma<128, 128, true><<<blocks, 256, 0, stream>>>(h1, pos, nbr, W1, b1, h2, N);
    gnn_layer_wmma<128, 16, false><<<blocks, 256, 0, stream>>>(h2, pos, nbr, W2, b2, out, N);
}